// ConvTranspose3d_26955214750400
// MI455X (gfx1250) — compile-verified
//
#include <hip/hip_runtime.h>

typedef __attribute__((ext_vector_type(2))) float v2f;
typedef __attribute__((ext_vector_type(8))) float v8f;

#define CI   128
#define CO   64
#define DI   16
#define HIN  32
#define WIN  32
#define DO_  32
#define HO   64
#define WO   64

#define XS_C (DI*HIN*WIN)   /* 16384  */
#define XS_N (CI*XS_C)      /* 2097152 */
#define OS_C (DO_*HO*WO)    /* 131072 */
#define OS_N (CO*OS_C)      /* 8388608 */

// Repack weight (ci,co,kd,kh,kw) -> w2[k64][ci/4][co][ci%4] so the two
// B-matrix K-values each lane needs are adjacent (one b64 load).
__global__ void pack_weights(const float* __restrict__ w, float* __restrict__ w2) {
    int idx = blockIdx.x * blockDim.x + threadIdx.x;
    if (idx >= CI * CO * 64) return;
    int cm  = idx & 3;
    int co  = (idx >> 2) & 63;
    int ci4 = (idx >> 8) & 31;
    int k   = idx >> 13;          // 0..63 flat (kd,kh,kw)
    int ci  = ci4 * 4 + cm;
    w2[idx] = w[(ci * CO + co) * 64 + k];
}

__device__ __forceinline__ v2f ld2(const float* p) { return *(const v2f*)p; }

// Block: 128 threads (4 wave32). Grid bits: [n:1][od:5][pair:5][half:1].
// pair -> two output rows oh0, oh0+2 (same parity => same kh0, shared B).
// Each block -> out[n, 0:64, od, {oh0,oh0+2}, 32*half : 32*half+32].
__global__ __launch_bounds__(128) void convT3d_wmma(
    const float* __restrict__ x, const float* __restrict__ w2,
    const float* __restrict__ bias, float* __restrict__ out) {

    // A tile: [td(2) x q(3) ih-planes][j(18)][ci(128), pad 132] = 57,024 B
    __shared__ float smem[6 * 18 * 132];

    const int bid    = blockIdx.x;
    const int half   = bid & 1;
    const int pair   = (bid >> 1) & 31;
    const int od     = (bid >> 6) & 31;
    const int n      = bid >> 11;
    const int parity = pair & 1;
    const int oh0    = parity + 4 * (pair >> 1);   // second row: oh0 + 2

    const int tid = threadIdx.x;
    const int id0 = (od + 1) >> 1;
    const int ih0 = (oh0 + 1) >> 1;                // row1 uses ih0 + 1
    const int ihb = ih0 - 1;                       // planes q=0,1,2 -> ih = ihb+q
    const int iwb = 16 * half - 1;                 // j in [0,18) -> iw = iwb + j

    // ---- stage A planes: smem[(td*3+q)*18 + j][ci] ----
    const float* xn = x + n * XS_N;
    for (int e = tid; e < 6 * 128 * 18; e += 128) {
        int j  = e % 18;                 // consecutive e -> consecutive iw (coalesced)
        int r  = e / 18;
        int ci = r & 127;
        int pl = r >> 7;                 // 0..5
        int td = pl / 3;
        int q  = pl % 3;
        int id = id0 - td;
        int ih = ihb + q;
        int iw = iwb + j;
        float v = 0.0f;
        if ((unsigned)id < (unsigned)DI && (unsigned)ih < (unsigned)HIN &&
            (unsigned)iw < (unsigned)WIN)
            v = xn[ci * XS_C + id * (HIN * WIN) + ih * WIN + iw];
        smem[(pl * 18 + j) * 132 + ci] = v;
    }
    __syncthreads();

    const int lane = tid & 31;
    const int wv   = tid >> 5;     // 0..3 -> co tile
    const int lm   = lane & 15;    // A: row M ; B: col N ; C: col N
    const int lh   = lane >> 4;    // K pair select (0,1)/(2,3); C row +8
    const int cob  = wv * 16;

    v8f acc00 = {0,0,0,0,0,0,0,0}; // row0, pw=0
    v8f acc01 = {0,0,0,0,0,0,0,0}; // row0, pw=1
    v8f acc10 = {0,0,0,0,0,0,0,0}; // row1, pw=0
    v8f acc11 = {0,0,0,0,0,0,0,0}; // row1, pw=1

    const int kd0 = (od  & 1) ? 0 : 1;
    const int kh0 = (oh0 & 1) ? 0 : 1;

    #pragma unroll
    for (int tt = 0; tt < 4; ++tt) {
        const int td    = tt >> 1;
        const int th    = tt & 1;
        const int kd    = kd0 + 2 * td;
        const int kh    = kh0 + 2 * th;
        const int kbase = (kd * 4 + kh) * 4;      // + kw -> flat k
        // row0 (oh0):   ih = ih0   - th -> plane q = 1 - th
        // row1 (oh0+2): ih = ih0+1 - th -> plane q = 2 - th
        const float* ar0 = &smem[((td * 3 + (1 - th)) * 18 + lm) * 132 + 2 * lh];
        const float* ar1 = &smem[((td * 3 + (2 - th)) * 18 + lm) * 132 + 2 * lh];
        const float* bb  = w2 + (cob + lm) * 4 + 2 * lh;
        #pragma unroll 2
        for (int cs = 0; cs < 32; ++cs) {
            const int ci = cs * 4;
            // A windows j = lm + joff, joff in {0,1,2}
            v2f a00 = ld2(ar0 + 0 * 132 + ci);
            v2f a01 = ld2(ar0 + 1 * 132 + ci);
            v2f a02 = ld2(ar0 + 2 * 132 + ci);
            v2f a10 = ld2(ar1 + 0 * 132 + ci);
            v2f a11 = ld2(ar1 + 1 * 132 + ci);
            v2f a12 = ld2(ar1 + 2 * 132 + ci);
            // B for kw = 0..3 — shared by both rows (one b64/lane, L2-resident)
            v2f b0 = ld2(bb + ((kbase + 0) * 32 + cs) * 256);
            v2f b1 = ld2(bb + ((kbase + 1) * 32 + cs) * 256);
            v2f b2 = ld2(bb + ((kbase + 2) * 32 + cs) * 256);
            v2f b3 = ld2(bb + ((kbase + 3) * 32 + cs) * 256);
            // pw=0: (tw=0 -> kw=1, joff=1) (tw=1 -> kw=3, joff=0)
            acc00 = __builtin_amdgcn_wmma_f32_16x16x4_f32(false, a01, false, b1,
                                                          (short)0, acc00, false, false);
            acc00 = __builtin_amdgcn_wmma_f32_16x16x4_f32(false, a00, false, b3,
                                                          (short)0, acc00, false, false);
            acc10 = __builtin_amdgcn_wmma_f32_16x16x4_f32(false, a11, false, b1,
                                                          (short)0, acc10, false, false);
            acc10 = __builtin_amdgcn_wmma_f32_16x16x4_f32(false, a10, false, b3,
                                                          (short)0, acc10, false, false);
            // pw=1: (tw=0 -> kw=0, joff=2) (tw=1 -> kw=2, joff=1)
            acc01 = __builtin_amdgcn_wmma_f32_16x16x4_f32(false, a02, false, b0,
                                                          (short)0, acc01, false, false);
            acc01 = __builtin_amdgcn_wmma_f32_16x16x4_f32(false, a01, false, b2,
                                                          (short)0, acc01, false, false);
            acc11 = __builtin_amdgcn_wmma_f32_16x16x4_f32(false, a12, false, b0,
                                                          (short)0, acc11, false, false);
            acc11 = __builtin_amdgcn_wmma_f32_16x16x4_f32(false, a11, false, b2,
                                                          (short)0, acc11, false, false);
        }
    }
    __syncthreads();

    // ---- transpose C through LDS: cl[row][co][owl], owl = pw + 2*(r+8*lh) ----
    float* cl = smem;                       // [2][64][33] = 4224 floats (fits)
    #pragma unroll
    for (int r = 0; r < 8; ++r) {
        int m = r + 8 * lh;
        cl[0 * (64 * 33) + (cob + lm) * 33 + 2 * m + 0] = acc00[r];
        cl[0 * (64 * 33) + (cob + lm) * 33 + 2 * m + 1] = acc01[r];
        cl[1 * (64 * 33) + (cob + lm) * 33 + 2 * m + 0] = acc10[r];
        cl[1 * (64 * 33) + (cob + lm) * 33 + 2 * m + 1] = acc11[r];
    }
    __syncthreads();

    // ---- coalesced stores (32 contiguous ow per 32 lanes) + bias ----
    float* outp = out + n * OS_N + od * (HO * WO) + oh0 * WO + 32 * half;
    for (int e = tid; e < 2 * CO * 32; e += 128) {
        int owl = e & 31;
        int co  = (e >> 5) & 63;
        int rw  = e >> 11;                  // 0 or 1 -> oh0, oh0+2
        outp[co * OS_C + rw * (2 * WO) + owl] = cl[rw * (64 * 33) + co * 33 + owl] + bias[co];
    }
}

extern "C" void kernel_launch(void* const* d_in, const int* in_sizes, int n_in,
                              void* d_out, int out_size, void* d_ws, size_t ws_size,
                              hipStream_t stream) {
    const float* x    = (const float*)d_in[0];
    const float* w    = (const float*)d_in[1];
    const float* bias = (const float*)d_in[2];
    float* out = (float*)d_out;
    float* w2  = (float*)d_ws;             // 2 MB packed weights

    pack_weights<<<dim3((CI * CO * 64) / 256), dim3(256), 0, stream>>>(w, w2);
    convT3d_wmma<<<dim3(2 * 32 * 32 * 2), dim3(128), 0, stream>>>(x, w2, bias, out);
}